// MambaLayer_63385127354409
// MI455X (gfx1250) — compile-verified
//
#include <hip/hip_runtime.h>
#include <math.h>

typedef float v2f __attribute__((ext_vector_type(2)));
typedef float v8f __attribute__((ext_vector_type(8)));

#define LDIM 32768       // 32*32*32
#define CDIM 96
#define EDIM 192         // D_INNER
#define TTOT 65536       // B*L tokens
#define NCH  64          // scan chunks
#define CLEN 512         // chunk length
#define EPSC 1e-5f

// ---------------- generic fp32 WMMA GEMM: Out(TxN) = A(TxK) * W(NxK)^T ----------
__global__ __launch_bounds__(32) void k_gemm_wmma(const float* __restrict__ A,
                                                  const float* __restrict__ W,
                                                  float* __restrict__ Out,
                                                  int K, int N) {
  int tok0 = blockIdx.x << 4;
  int col0 = blockIdx.y << 4;
  int lane = threadIdx.x;
  int lo = lane & 15, hi = lane >> 4;
  const float* ap = A + (size_t)(tok0 + lo) * K + 2 * hi;
  const float* wp = W + (size_t)(col0 + lo) * K + 2 * hi;
  __builtin_prefetch(ap + (size_t)16 * K, 0, 1);
  v8f c = {};
  for (int k0 = 0; k0 < K; k0 += 4) {
    v2f a = *(const v2f*)(ap + k0);
    v2f b = *(const v2f*)(wp + k0);
    c = __builtin_amdgcn_wmma_f32_16x16x4_f32(false, a, false, b, (short)0, c, false, false);
  }
  size_t base = (size_t)(tok0 + 8 * hi) * N + col0 + lo;
#pragma unroll
  for (int v = 0; v < 8; ++v) Out[base + (size_t)v * N] = c[v];
}

// ---------------- per-(b,c) mean / rstd over 32768 elems ------------------------
__global__ __launch_bounds__(256) void k_stats(const float* __restrict__ src,
                                               float* __restrict__ stats, int mode) {
  int bc = blockIdx.x;                       // 0..383
  size_t base; int stride;
  if (mode == 0) { base = (size_t)bc * LDIM; stride = 1; }
  else { int b = bc / CDIM, c = bc % CDIM; base = (size_t)b * LDIM * CDIM + c; stride = CDIM; }
  float s = 0.f, sq = 0.f;
  for (int i = threadIdx.x; i < LDIM; i += 256) {
    float v = src[base + (size_t)i * stride];
    s += v; sq += v * v;
  }
  __shared__ float ls[256], lq[256];
  ls[threadIdx.x] = s; lq[threadIdx.x] = sq;
  __syncthreads();
  for (int off = 128; off > 0; off >>= 1) {
    if (threadIdx.x < off) { ls[threadIdx.x] += ls[threadIdx.x + off]; lq[threadIdx.x] += lq[threadIdx.x + off]; }
    __syncthreads();
  }
  if (threadIdx.x == 0) {
    float m = ls[0] / (float)LDIM;
    float var = lq[0] / (float)LDIM - m * m;
    stats[bc * 2] = m;
    stats[bc * 2 + 1] = rsqrtf(var + EPSC);
  }
}

// ---------------- seq[(b*L+l)*96+c] = inorm(x) ----------------------------------
__global__ __launch_bounds__(256) void k_seq(const float* __restrict__ x,
                                             const float* __restrict__ st,
                                             float* __restrict__ seq) {
  size_t gid = (size_t)blockIdx.x * 256 + threadIdx.x;   // (b*96+c)*32768 + l
  int bc = (int)(gid >> 15), l = (int)(gid & 32767);
  int b = bc / CDIM, c = bc % CDIM;
  float v = (x[gid] - st[bc * 2]) * st[bc * 2 + 1];
  seq[((size_t)b * LDIM + l) * CDIM + c] = v;
}

// ---------------- SPPE: directional pooling -------------------------------------
__global__ __launch_bounds__(256) void k_pool(const float* __restrict__ x,
                                              float* __restrict__ pooled) {
  int ch = blockIdx.x;                       // bg*12+cg
  int bg = ch / 12, cg = ch % 12;
  int b = bg >> 3, g = bg & 7;
  const float* base = x + (size_t)(b * CDIM + g * 12 + cg) * LDIM;
  int wv = threadIdx.x >> 5, lane = threadIdx.x & 31;
  for (int s = wv; s < 96; s += 8) {
    float sum = 0.f;
    if (s < 32) {
      const float* p = base + s * 1024;
      for (int i = lane; i < 1024; i += 32) sum += p[i];
    } else if (s < 64) {
      int ww = s - 32;
      for (int j = lane; j < 1024; j += 32) { int h = j >> 5, d = j & 31; sum += base[h * 1024 + ww * 32 + d]; }
    } else {
      int dd = s - 64;
      for (int j = lane; j < 1024; j += 32) { int h = j >> 5, w2 = j & 31; sum += base[h * 1024 + w2 * 32 + dd]; }
    }
    sum += __shfl_xor(sum, 1, 32); sum += __shfl_xor(sum, 2, 32);
    sum += __shfl_xor(sum, 4, 32); sum += __shfl_xor(sum, 8, 32);
    sum += __shfl_xor(sum, 16, 32);
    if (lane == 0) pooled[(size_t)ch * 96 + s] = sum * (1.f / 1024.f);
  }
}

// ---------------- SPPE: 12x12 conv1 + sigmoid gates -----------------------------
__global__ __launch_bounds__(256) void k_gates(const float* __restrict__ pooled,
                                               const float* __restrict__ c1w,
                                               const float* __restrict__ c1b,
                                               float* __restrict__ gates) {
  int gid = blockIdx.x * 256 + threadIdx.x;  // (bg*12+o)*96 + pos ; total 18432
  int pos = gid % 96, rest = gid / 96;
  int o = rest % 12, bg = rest / 12;
  float acc = c1b[o];
#pragma unroll
  for (int ci = 0; ci < 12; ++ci) acc += c1w[o * 12 + ci] * pooled[(size_t)(bg * 12 + ci) * 96 + pos];
  gates[gid] = 1.f / (1.f + expf(-acc));
}

// ---------------- SPPE: gated mean/var per (bg,cg) ------------------------------
__global__ __launch_bounds__(256) void k_gstats(const float* __restrict__ x,
                                                const float* __restrict__ gates,
                                                float* __restrict__ gst) {
  int ch = blockIdx.x;
  int bg = ch / 12, cg = ch % 12;
  int b = bg >> 3, g = bg & 7;
  const float* base = x + (size_t)(b * CDIM + g * 12 + cg) * LDIM;
  const float* gt = gates + (size_t)ch * 96;
  float s = 0.f, sq = 0.f;
  for (int n = threadIdx.x; n < LDIM; n += 256) {
    int h = n >> 10, w2 = (n >> 5) & 31, d = n & 31;
    float gv = base[n] * gt[h] * gt[32 + w2] * gt[64 + d];
    s += gv; sq += gv * gv;
  }
  __shared__ float ls[256], lq[256];
  ls[threadIdx.x] = s; lq[threadIdx.x] = sq;
  __syncthreads();
  for (int off = 128; off > 0; off >>= 1) {
    if (threadIdx.x < off) { ls[threadIdx.x] += ls[threadIdx.x + off]; lq[threadIdx.x] += lq[threadIdx.x + off]; }
    __syncthreads();
  }
  if (threadIdx.x == 0) {
    float m = ls[0] / (float)LDIM;
    float var = lq[0] / (float)LDIM - m * m;
    gst[ch * 2] = m;
    gst[ch * 2 + 1] = rsqrtf(var + EPSC);
  }
}

// ---------------- SPPE: 3x3x3 group conv + per-block partial sums ---------------
__global__ __launch_bounds__(256) void k_conv3(const float* __restrict__ x,
                                               const float* __restrict__ c3w,
                                               const float* __restrict__ c3b,
                                               float* __restrict__ x2,
                                               float* __restrict__ x2part) {
  int bid = blockIdx.x;                      // ch*128 + chunk
  int chunk = bid & 127, ch = bid >> 7;
  int bg = ch / 12, co = ch % 12;
  int b = bg >> 3, g = bg & 7;
  __shared__ float wl[324];
  if (threadIdx.x < 324) wl[threadIdx.x] = c3w[co * 324 + threadIdx.x];
  __syncthreads();
  int n = chunk * 256 + threadIdx.x;
  int h = n >> 10, w2 = (n >> 5) & 31, d = n & 31;
  float acc = c3b[co];
  const float* xg = x + (size_t)(b * CDIM + g * 12) * LDIM;
  for (int ci = 0; ci < 12; ++ci) {
    const float* xcib = xg + (size_t)ci * LDIM;
    const float* wp = wl + ci * 27;
    for (int kh = 0; kh < 3; ++kh) {
      int hh = h + kh - 1; if ((unsigned)hh >= 32u) continue;
      for (int kw = 0; kw < 3; ++kw) {
        int w3 = w2 + kw - 1; if ((unsigned)w3 >= 32u) continue;
        for (int kd = 0; kd < 3; ++kd) {
          int d3 = d + kd - 1; if ((unsigned)d3 >= 32u) continue;
          acc += xcib[hh * 1024 + w3 * 32 + d3] * wp[kh * 9 + kw * 3 + kd];
        }
      }
    }
  }
  x2[(size_t)ch * LDIM + n] = acc;
  __shared__ float ls[256];
  ls[threadIdx.x] = acc;
  __syncthreads();
  for (int off = 128; off > 0; off >>= 1) {
    if (threadIdx.x < off) ls[threadIdx.x] += ls[threadIdx.x + off];
    __syncthreads();
  }
  if (threadIdx.x == 0) x2part[(size_t)ch * 128 + chunk] = ls[0];
}

// ---------------- SPPE: reduce partials to x2 means -----------------------------
__global__ __launch_bounds__(256) void k_x2red(const float* __restrict__ x2part,
                                               float* __restrict__ x2mean) {
  int ch = threadIdx.x;
  if (ch < 192) {
    float s = 0.f;
    for (int c = 0; c < 128; ++c) s += x2part[(size_t)ch * 128 + c];
    x2mean[ch] = s / (float)LDIM;
  }
}

// ---------------- SPPE: softmaxes (x11 from gn_b, x21 from x2 means) ------------
__global__ __launch_bounds__(32) void k_smax(const float* __restrict__ x2mean,
                                             const float* __restrict__ gnb,
                                             float* __restrict__ x11,
                                             float* __restrict__ x21) {
  int tid = threadIdx.x;
  if (tid < 16) {
    int bg = tid;
    float mx = -1e30f;
    for (int c = 0; c < 12; ++c) mx = fmaxf(mx, x2mean[bg * 12 + c]);
    float sum = 0.f, e[12];
    for (int c = 0; c < 12; ++c) { e[c] = expf(x2mean[bg * 12 + c] - mx); sum += e[c]; }
    for (int c = 0; c < 12; ++c) x21[bg * 12 + c] = e[c] / sum;
  } else if (tid == 16) {
    float mx = -1e30f;
    for (int c = 0; c < 12; ++c) mx = fmaxf(mx, gnb[c]);
    float sum = 0.f, e[12];
    for (int c = 0; c < 12; ++c) { e[c] = expf(gnb[c] - mx); sum += e[c]; }
    for (int c = 0; c < 12; ++c) x11[c] = e[c] / sum;
  }
}

// ---------------- SPPE: weights + y_sppe ----------------------------------------
__global__ __launch_bounds__(256) void k_sppe_out(const float* __restrict__ x,
                                                  const float* __restrict__ gates,
                                                  const float* __restrict__ gst,
                                                  const float* __restrict__ gnw,
                                                  const float* __restrict__ gnb,
                                                  const float* __restrict__ x2,
                                                  const float* __restrict__ x11,
                                                  const float* __restrict__ x21,
                                                  float* __restrict__ ysppe) {
  int bid = blockIdx.x;                      // bg*128 + chunk
  int chunk = bid & 127, bg = bid >> 7;
  int b = bg >> 3, g = bg & 7;
  int n = chunk * 256 + threadIdx.x;
  int h = n >> 10, w2 = (n >> 5) & 31, d = n & 31;
  float xv[12];
  float wsum = 0.f;
#pragma unroll
  for (int c = 0; c < 12; ++c) {
    int ch = bg * 12 + c;
    const float* gt = gates + (size_t)ch * 96;
    float xval = x[(size_t)(b * CDIM + g * 12 + c) * LDIM + n];
    xv[c] = xval;
    float gated = xval * gt[h] * gt[32 + w2] * gt[64 + d];
    float x1v = (gated - gst[ch * 2]) * gst[ch * 2 + 1] * gnw[c] + gnb[c];
    float x2v = x2[(size_t)ch * LDIM + n];
    wsum += x11[c] * x2v + x21[ch] * x1v;
  }
  float sig = 1.f / (1.f + expf(-wsum));
#pragma unroll
  for (int c = 0; c < 12; ++c)
    ysppe[(size_t)(b * CDIM + g * 12 + c) * LDIM + n] = xv[c] * sig;
}

// ---------------- depthwise causal conv1d (k=4) + silu --------------------------
__global__ __launch_bounds__(256) void k_conv1d(const float* __restrict__ xz,
                                                const float* __restrict__ w,
                                                const float* __restrict__ bia,
                                                float* __restrict__ xc) {
  size_t gid = (size_t)blockIdx.x * 256 + threadIdx.x;  // (b*L+l)*192 + e
  int e = (int)(gid % EDIM);
  size_t tl = gid / EDIM;
  int l = (int)(tl & 32767), b = (int)(tl >> 15);
  float acc = bia[e];
  const float* wp = w + e * 4;
#pragma unroll
  for (int k = 0; k < 4; ++k) {
    int li = l - 3 + k;
    if (li >= 0) acc += xz[((size_t)b * LDIM + li) * 384 + e] * wp[k];
  }
  acc = acc / (1.f + expf(-acc));
  xc[gid] = acc;
}

// ---------------- x_proj (scalar, K=192, N=38) ----------------------------------
__global__ __launch_bounds__(256) void k_xproj(const float* __restrict__ xc,
                                               const float* __restrict__ wpj,
                                               float* __restrict__ dbl) {
  size_t gid = (size_t)blockIdx.x * 256 + threadIdx.x;  // t*38 + o
  int o = (int)(gid % 38);
  size_t t = gid / 38;
  const float* xr = xc + t * EDIM;
  const float* wr = wpj + (size_t)o * EDIM;
  float acc = 0.f;
  for (int k = 0; k < EDIM; ++k) acc += xr[k] * wr[k];
  dbl[gid] = acc;
}

// ---------------- dt_proj + softplus --------------------------------------------
__global__ __launch_bounds__(256) void k_dt(const float* __restrict__ dbl,
                                            const float* __restrict__ dtw,
                                            const float* __restrict__ dtb,
                                            float* __restrict__ dtbuf) {
  size_t gid = (size_t)blockIdx.x * 256 + threadIdx.x;  // t*192 + e
  int e = (int)(gid % EDIM);
  size_t t = gid / EDIM;
  float acc = dtb[e];
#pragma unroll
  for (int r = 0; r < 6; ++r) acc += dbl[t * 38 + r] * dtw[e * 6 + r];
  acc = (acc > 20.f) ? acc : log1pf(expf(acc));
  dtbuf[gid] = acc;
}

// ---------------- scan pass 1: per-chunk local scan -----------------------------
__global__ __launch_bounds__(32) void k_scan1(const float* __restrict__ dtb_,
                                              const float* __restrict__ xc,
                                              const float* __restrict__ dbl,
                                              const float* __restrict__ alog,
                                              float* __restrict__ ys,
                                              float* __restrict__ hfin,
                                              float* __restrict__ Pc) {
  int idx = blockIdx.x;                      // b*(192*64) + e*64 + c
  int c = idx & 63;
  int e = (idx >> 6) % EDIM;
  int b = idx / (64 * EDIM);
  int lane = threadIdx.x, n = lane & 15;
  float An = -expf(alog[e * 16 + n]);
  float h = 0.f, cum = 1.f;
  size_t t0 = (size_t)b * LDIM + (size_t)c * CLEN;
  for (int tt = 0; tt < CLEN; ++tt) {
    size_t t = t0 + tt;
    float dtv = dtb_[t * EDIM + e];
    float xv  = xc[t * EDIM + e];
    float a = expf(dtv * An);
    float Bv = dbl[t * 38 + 6 + n];
    float Cv = dbl[t * 38 + 22 + n];
    h = h * a + dtv * xv * Bv;
    cum *= a;
    float p = h * Cv;
    p += __shfl_xor(p, 1, 32); p += __shfl_xor(p, 2, 32);
    p += __shfl_xor(p, 4, 32); p += __shfl_xor(p, 8, 32);
    if (lane == 0) ys[t * EDIM + e] = p;
  }
  if (lane < 16) {
    size_t o = (((size_t)(b * EDIM + e) * NCH) + c) * 16 + n;
    hfin[o] = h; Pc[o] = cum;
  }
}

// ---------------- scan pass 2: chunk-prefix -------------------------------------
__global__ __launch_bounds__(32) void k_scan2(const float* __restrict__ hfin,
                                              const float* __restrict__ Pc,
                                              float* __restrict__ hstart) {
  int be = blockIdx.x;                       // 0..383
  int lane = threadIdx.x;
  if (lane >= 16) return;
  float hs = 0.f;
  for (int c = 0; c < NCH; ++c) {
    size_t o = ((size_t)be * NCH + c) * 16 + lane;
    hstart[o] = hs;
    hs = hs * Pc[o] + hfin[o];
  }
}

// ---------------- scan pass 3: decay-weighted fixup -----------------------------
__global__ __launch_bounds__(32) void k_scan3(const float* __restrict__ dtb_,
                                              const float* __restrict__ dbl,
                                              const float* __restrict__ alog,
                                              const float* __restrict__ hstart,
                                              float* __restrict__ ys) {
  int idx = blockIdx.x;
  int c = idx & 63;
  if (c == 0) return;
  int e = (idx >> 6) % EDIM;
  int b = idx / (64 * EDIM);
  int lane = threadIdx.x, n = lane & 15;
  float An = -expf(alog[e * 16 + n]);
  float hs = hstart[(((size_t)(b * EDIM + e) * NCH) + c) * 16 + n];
  float cum = 1.f;
  size_t t0 = (size_t)b * LDIM + (size_t)c * CLEN;
  for (int tt = 0; tt < CLEN; ++tt) {
    size_t t = t0 + tt;
    float dtv = dtb_[t * EDIM + e];
    float a = expf(dtv * An);
    cum *= a;
    float Cv = dbl[t * 38 + 22 + n];
    float p = hs * cum * Cv;
    p += __shfl_xor(p, 1, 32); p += __shfl_xor(p, 2, 32);
    p += __shfl_xor(p, 4, 32); p += __shfl_xor(p, 8, 32);
    if (lane == 0) ys[t * EDIM + e] += p;
  }
}

// ---------------- y = (ys + xc*D) * silu(zg) ------------------------------------
__global__ __launch_bounds__(256) void k_y(const float* __restrict__ xz,
                                           const float* __restrict__ xc,
                                           const float* __restrict__ ssd,
                                           float* __restrict__ ys) {
  size_t gid = (size_t)blockIdx.x * 256 + threadIdx.x;
  int e = (int)(gid % EDIM);
  size_t tl = gid / EDIM;
  float z = xz[tl * 384 + EDIM + e];
  float sil = z / (1.f + expf(-z));
  ys[gid] = (ys[gid] + xc[gid] * ssd[e]) * sil;
}

// ---------------- xo = inorm(out) * y_sppe (in place over ysppe) ----------------
__global__ __launch_bounds__(256) void k_xo(const float* __restrict__ outb,
                                            const float* __restrict__ st,
                                            float* __restrict__ ysppe) {
  size_t gid = (size_t)blockIdx.x * 256 + threadIdx.x;   // (b*96+c)*32768 + n
  int bc = (int)(gid >> 15), n = (int)(gid & 32767);
  int b = bc / CDIM, c = bc % CDIM;
  float v = (outb[((size_t)b * LDIM + n) * CDIM + c] - st[bc * 2]) * st[bc * 2 + 1];
  ysppe[gid] *= v;
}

// ---------------- fused MLP: fc1 -> gelu -> fc2 -> +x (WMMA, LDS-staged) --------
__global__ __launch_bounds__(256) void k_mlp(const float* __restrict__ xo,
                                             const float* __restrict__ w1,
                                             const float* __restrict__ b1,
                                             const float* __restrict__ w2,
                                             const float* __restrict__ b2,
                                             const float* __restrict__ xin,
                                             float* __restrict__ out) {
  __shared__ float tls[384 * 16];
  int tile = blockIdx.x;                     // 0..4095
  int b = tile >> 11;
  int vox0 = (tile & 2047) << 4;
  int tid = threadIdx.x;
  int wv = tid >> 5, lane = tid & 31;
  int lo = lane & 15, hi = lane >> 4;
  size_t xobase = (size_t)b * CDIM * LDIM + vox0 + lo;
  // stage 1: t(384x16) = gelu(W1(384x96) * xo(96x16) + b1)
  for (int mt = wv; mt < 24; mt += 8) {
    int m0 = mt << 4;
    const float* w1p = w1 + (size_t)(m0 + lo) * CDIM + 2 * hi;
    v8f acc = {};
    for (int k0 = 0; k0 < CDIM; k0 += 4) {
      v2f a = *(const v2f*)(w1p + k0);
      int kk = k0 + 2 * hi;
      v2f bb;
      bb.x = xo[xobase + (size_t)kk * LDIM];
      bb.y = xo[xobase + (size_t)(kk + 1) * LDIM];
      acc = __builtin_amdgcn_wmma_f32_16x16x4_f32(false, a, false, bb, (short)0, acc, false, false);
    }
#pragma unroll
    for (int v = 0; v < 8; ++v) {
      int m = m0 + v + 8 * hi;
      float val = acc[v] + b1[m];
      val = 0.5f * val * (1.f + erff(val * 0.70710678118654752f));
      tls[m * 16 + lo] = val;
    }
  }
  __syncthreads();
  // stage 2: out(96x16) = W2(96x384) * t(384x16) + b2 + x
  if (wv < 6) {
    int m0 = wv << 4;
    const float* w2p = w2 + (size_t)(m0 + lo) * 384 + 2 * hi;
    v8f acc = {};
    for (int k0 = 0; k0 < 384; k0 += 4) {
      v2f a = *(const v2f*)(w2p + k0);
      int kk = k0 + 2 * hi;
      v2f bb;
      bb.x = tls[kk * 16 + lo];
      bb.y = tls[(kk + 1) * 16 + lo];
      acc = __builtin_amdgcn_wmma_f32_16x16x4_f32(false, a, false, bb, (short)0, acc, false, false);
    }
#pragma unroll
    for (int v = 0; v < 8; ++v) {
      int m = m0 + v + 8 * hi;
      size_t oidx = ((size_t)b * CDIM + m) * LDIM + vox0 + lo;
      out[oidx] = acc[v] + b2[m] + xin[oidx];
    }
  }
}

extern "C" void kernel_launch(void* const* d_in, const int* in_sizes, int n_in,
                              void* d_out, int out_size, void* d_ws, size_t ws_size,
                              hipStream_t stream) {
  (void)in_sizes; (void)n_in; (void)out_size; (void)ws_size;
  const float* x    = (const float*)d_in[0];
  const float* c1w  = (const float*)d_in[1];
  const float* c1b  = (const float*)d_in[2];
  const float* c3w  = (const float*)d_in[3];
  const float* c3b  = (const float*)d_in[4];
  const float* gnw  = (const float*)d_in[5];
  const float* gnb  = (const float*)d_in[6];
  const float* inpw = (const float*)d_in[7];
  const float* cv1w = (const float*)d_in[8];
  const float* cv1b = (const float*)d_in[9];
  const float* xpw  = (const float*)d_in[10];
  const float* dtw  = (const float*)d_in[11];
  const float* dtb  = (const float*)d_in[12];
  const float* alog = (const float*)d_in[13];
  const float* ssd  = (const float*)d_in[14];
  const float* opw  = (const float*)d_in[15];
  const float* w1   = (const float*)d_in[16];
  const float* b1   = (const float*)d_in[17];
  const float* w2   = (const float*)d_in[18];
  const float* b2   = (const float*)d_in[19];
  float* out = (float*)d_out;
  float* ws  = (float*)d_ws;

  size_t off = 0;
  float* f_seq   = ws + off; off += (size_t)TTOT * CDIM;     // 6.29M
  float* f_xz    = ws + off; off += (size_t)TTOT * 384;      // 25.2M
  float* f_xc    = ws + off; off += (size_t)TTOT * EDIM;     // 12.6M
  float* f_dbl   = ws + off; off += (size_t)TTOT * 38;       // 2.5M
  float* f_dt    = ws + off; off += (size_t)TTOT * EDIM;     // 12.6M
  float* f_ys    = ws + off; off += (size_t)TTOT * EDIM;     // 12.6M (ys -> y)
  float* f_out   = ws + off; off += (size_t)TTOT * CDIM;     // 6.29M
  float* f_ysppe = ws + off; off += (size_t)2 * CDIM * LDIM; // 6.29M (ysppe -> xo)
  float* f_x2    = ws + off; off += (size_t)192 * LDIM;      // 6.29M
  float* f_pool  = ws + off; off += 192 * 96;
  float* f_gates = ws + off; off += 192 * 96;
  float* f_st1   = ws + off; off += 768;
  float* f_st2   = ws + off; off += 768;
  float* f_gst   = ws + off; off += 384;
  float* f_x2pt  = ws + off; off += 192 * 128;
  float* f_x2m   = ws + off; off += 192;
  float* f_x11   = ws + off; off += 16;
  float* f_x21   = ws + off; off += 192;
  float* f_hfin  = ws + off; off += (size_t)384 * NCH * 16;
  float* f_P     = ws + off; off += (size_t)384 * NCH * 16;
  float* f_hst   = ws + off; off += (size_t)384 * NCH * 16;

  // ---- instance norm of x -> seq (B,L,C)
  k_stats<<<384, 256, 0, stream>>>(x, f_st1, 0);
  k_seq<<<24576, 256, 0, stream>>>(x, f_st1, f_seq);

  // ---- SPPE branch
  k_pool<<<192, 256, 0, stream>>>(x, f_pool);
  k_gates<<<72, 256, 0, stream>>>(f_pool, c1w, c1b, f_gates);
  k_gstats<<<192, 256, 0, stream>>>(x, f_gates, f_gst);
  k_conv3<<<24576, 256, 0, stream>>>(x, c3w, c3b, f_x2, f_x2pt);
  k_x2red<<<1, 256, 0, stream>>>(f_x2pt, f_x2m);
  k_smax<<<1, 32, 0, stream>>>(f_x2m, gnb, f_x11, f_x21);
  k_sppe_out<<<2048, 256, 0, stream>>>(x, f_gates, f_gst, gnw, gnb, f_x2, f_x11, f_x21, f_ysppe);

  // ---- Mamba: in_proj (WMMA), conv1d+silu, x_proj, dt_proj
  k_gemm_wmma<<<dim3(TTOT / 16, 384 / 16), 32, 0, stream>>>(f_seq, inpw, f_xz, CDIM, 384);
  k_conv1d<<<49152, 256, 0, stream>>>(f_xz, cv1w, cv1b, f_xc);
  k_xproj<<<9728, 256, 0, stream>>>(f_xc, xpw, f_dbl);
  k_dt<<<49152, 256, 0, stream>>>(f_dbl, dtw, dtb, f_dt);

  // ---- chunked selective scan (exact 3-pass)
  k_scan1<<<2 * EDIM * NCH, 32, 0, stream>>>(f_dt, f_xc, f_dbl, alog, f_ys, f_hfin, f_P);
  k_scan2<<<384, 32, 0, stream>>>(f_hfin, f_P, f_hst);
  k_scan3<<<2 * EDIM * NCH, 32, 0, stream>>>(f_dt, f_dbl, alog, f_hst, f_ys);

  // ---- gate + out_proj (WMMA)
  k_y<<<49152, 256, 0, stream>>>(f_xz, f_xc, ssd, f_ys);
  k_gemm_wmma<<<dim3(TTOT / 16, CDIM / 16), 32, 0, stream>>>(f_ys, opw, f_out, EDIM, CDIM);

  // ---- inorm(out) * y_sppe -> xo (in place), then fused MLP + residual (WMMA)
  k_stats<<<384, 256, 0, stream>>>(f_out, f_st2, 1);
  k_xo<<<24576, 256, 0, stream>>>(f_out, f_st2, f_ysppe);
  k_mlp<<<4096, 256, 0, stream>>>(f_ysppe, w1, b1, w2, b2, x, out);
}